// ImprovedGCNRegressor_83451214562002
// MI455X (gfx1250) — compile-verified
//
#include <hip/hip_runtime.h>
#include <hip/hip_bf16.h>

#define N_NODES 100000
#define N_EDGES 1600000
#define C 128
#define CH 64
#define L 4
#define EPS 1e-5f
#define SMS 132   // LDS row stride (floats): 132 % 64 banks = 4 -> conflict-free column reads
#define RT 4      // row tiles (of 16 nodes) per GEMM block -> 4x register reuse of B operand

typedef float v2f __attribute__((ext_vector_type(2)));
typedef float v8f __attribute__((ext_vector_type(8)));

// ---------------- degree / inverse count ----------------
__global__ void degree_kernel(const long long* __restrict__ dst, float* __restrict__ cnt, int E) {
    int e = blockIdx.x * blockDim.x + threadIdx.x;
    if (e < E) atomicAdd(&cnt[(int)dst[e]], 1.0f);
}

__global__ void inv_cnt_kernel(float* __restrict__ c, int n) {
    int i = blockIdx.x * blockDim.x + threadIdx.x;
    if (i < n) c[i] = 1.0f / fmaxf(c[i], 1.0f);
}

// ---------------- edge scatter: one wave32 per edge, float4 per lane ----------------
__global__ __launch_bounds__(256) void scatter_kernel(const float* __restrict__ h,
                                                      const long long* __restrict__ src,
                                                      const long long* __restrict__ dst,
                                                      float* __restrict__ agg, int E) {
    int gid  = blockIdx.x * blockDim.x + threadIdx.x;
    int e    = gid >> 5;
    int lane = gid & 31;
    if (e >= E) return;
    int s = (int)src[e];
    int d = (int)dst[e];
    const float4 v = *(const float4*)(h + (size_t)s * C + lane * 4);
    float* o = agg + (size_t)d * C + lane * 4;
    atomicAdd(o + 0, v.x);
    atomicAdd(o + 1, v.y);
    atomicAdd(o + 2, v.z);
    atomicAdd(o + 3, v.w);
}

// ---------------- fused SAGE GEMM: out = relu(bn(mean@Wl + h@Wr + b)) [+res] ----------------
// grid.x = ceil(N/(16*RT)); 256 threads = 8 waves, wave w -> output cols [16w,16w+16)
// k-loop outer: B (weights) loaded once per k-step, reused across RT row tiles in registers.
__global__ __launch_bounds__(256) void sage_gemm_kernel(
    const float* __restrict__ agg, const float* __restrict__ invc,
    const float* __restrict__ h,
    const float* __restrict__ Wl, const float* __restrict__ Wr,
    const float* __restrict__ bias,
    const float* __restrict__ bn_g, const float* __restrict__ bn_b,
    const float* __restrict__ bn_m, const float* __restrict__ bn_v,
    const float* __restrict__ residual,   // nullptr if no residual
    float* __restrict__ out) {
    __shared__ float sM[16 * RT * SMS];
    __shared__ float sH[16 * RT * SMS];

    const int tid   = threadIdx.x;
    const int node0 = blockIdx.x * (16 * RT);

    // Stage (16*RT)x128 tiles of mean (= agg * invc) and h into LDS; zero-fill tail rows
    for (int i = tid; i < 16 * RT * 32; i += 256) {
        int r    = i >> 5;
        int c4   = (i & 31) * 4;
        int node = node0 + r;
        float4 vm = {0.f, 0.f, 0.f, 0.f};
        float4 vh = {0.f, 0.f, 0.f, 0.f};
        if (node < N_NODES) {
            float ic = invc[node];
            float4 va = *(const float4*)(agg + (size_t)node * C + c4);
            vh = *(const float4*)(h + (size_t)node * C + c4);
            vm.x = va.x * ic; vm.y = va.y * ic; vm.z = va.z * ic; vm.w = va.w * ic;
        }
        sM[r * SMS + c4 + 0] = vm.x;
        sM[r * SMS + c4 + 1] = vm.y;
        sM[r * SMS + c4 + 2] = vm.z;
        sM[r * SMS + c4 + 3] = vm.w;
        sH[r * SMS + c4 + 0] = vh.x;
        sH[r * SMS + c4 + 1] = vh.y;
        sH[r * SMS + c4 + 2] = vh.z;
        sH[r * SMS + c4 + 3] = vh.w;
    }
    __syncthreads();

    const int wave = tid >> 5;
    const int lane = tid & 31;
    const int row  = lane & 15;               // A row within a 16-row tile
    const int col  = wave * 16 + (lane & 15); // global output column
    const int khi  = (lane >> 4) << 1;        // lanes 0-15 -> K 0,1 ; lanes 16-31 -> K 2,3

    v8f accL[RT];
    v8f accR[RT];
#pragma unroll
    for (int rt = 0; rt < RT; ++rt) { accL[rt] = (v8f){}; accR[rt] = (v8f){}; }

#pragma unroll
    for (int k0 = 0; k0 < C; k0 += 4) {
        const int kk = k0 + khi;
        v2f bL, bR;
        bL.x = Wl[kk * C + col];
        bL.y = Wl[(kk + 1) * C + col];
        bR.x = Wr[kk * C + col];
        bR.y = Wr[(kk + 1) * C + col];
#pragma unroll
        for (int rt = 0; rt < RT; ++rt) {
            const int rr = rt * 16 + row;
            v2f aM, aH;
            aM.x = sM[rr * SMS + kk];
            aM.y = sM[rr * SMS + kk + 1];
            aH.x = sH[rr * SMS + kk];
            aH.y = sH[rr * SMS + kk + 1];
            accL[rt] = __builtin_amdgcn_wmma_f32_16x16x4_f32(false, aM, false, bL,
                                                             (short)0, accL[rt], false, false);
            accR[rt] = __builtin_amdgcn_wmma_f32_16x16x4_f32(false, aH, false, bR,
                                                             (short)0, accR[rt], false, false);
        }
    }

    // Fused epilogue: bias -> BN(eval) -> ReLU -> optional residual
    const float scale = bn_g[col] * __frsqrt_rn(bn_v[col] + EPS);
    const float shift = bn_b[col] - bn_m[col] * scale;
    const float bb    = bias[col];
    const int jbase   = (lane >> 4) << 3;     // +8 for upper half-lanes
#pragma unroll
    for (int rt = 0; rt < RT; ++rt) {
#pragma unroll
        for (int j = 0; j < 8; ++j) {
            int r = node0 + rt * 16 + jbase + j;
            if (r < N_NODES) {
                float xx = accL[rt][j] + accR[rt][j] + bb;
                float y  = fmaf(xx, scale, shift);
                y = fmaxf(y, 0.0f);
                if (residual) y += residual[(size_t)r * C + col];
                out[(size_t)r * C + col] = y;
            }
        }
    }
}

// ---------------- LayerNorm over feature dim: one wave32 per node ----------------
__global__ __launch_bounds__(256) void layernorm_kernel(const float* __restrict__ in,
                                                        const float* __restrict__ g,
                                                        const float* __restrict__ b,
                                                        float* __restrict__ out, int n) {
    int gid  = blockIdx.x * blockDim.x + threadIdx.x;
    int node = gid >> 5;
    int lane = gid & 31;
    if (node >= n) return;
    float4 v = *(const float4*)(in + (size_t)node * C + lane * 4);
    float s = v.x + v.y + v.z + v.w;
    float q = v.x * v.x + v.y * v.y + v.z * v.z + v.w * v.w;
    for (int m = 16; m; m >>= 1) {
        s += __shfl_xor(s, m, 32);
        q += __shfl_xor(q, m, 32);
    }
    float mu  = s * (1.0f / 128.0f);
    float var = q * (1.0f / 128.0f) - mu * mu;
    float rs  = __frsqrt_rn(var + EPS);
    int c = lane * 4;
    float4 o;
    o.x = (v.x - mu) * rs * g[c + 0] + b[c + 0];
    o.y = (v.y - mu) * rs * g[c + 1] + b[c + 1];
    o.z = (v.z - mu) * rs * g[c + 2] + b[c + 2];
    o.w = (v.w - mu) * rs * g[c + 3] + b[c + 3];
    *(float4*)(out + (size_t)node * C + c) = o;
}

// ---------------- head lin1 (128->64) + BN + ReLU via WMMA: 4 waves per 16-row tile ----------------
__global__ __launch_bounds__(128) void head1_kernel(const float* __restrict__ hin,
                                                    const float* __restrict__ W1,
                                                    const float* __restrict__ b1,
                                                    const float* __restrict__ g,
                                                    const float* __restrict__ bb,
                                                    const float* __restrict__ m,
                                                    const float* __restrict__ v,
                                                    float* __restrict__ out) {
    __shared__ float sA[16 * SMS];
    const int tid   = threadIdx.x;
    const int node0 = blockIdx.x * 16;
    for (int i = tid; i < 16 * 32; i += 128) {
        int r  = i >> 5;
        int c4 = (i & 31) * 4;
        float4 va = *(const float4*)(hin + (size_t)(node0 + r) * C + c4);
        sA[r * SMS + c4 + 0] = va.x;
        sA[r * SMS + c4 + 1] = va.y;
        sA[r * SMS + c4 + 2] = va.z;
        sA[r * SMS + c4 + 3] = va.w;
    }
    __syncthreads();

    const int wave = tid >> 5;
    const int lane = tid & 31;
    const int row  = lane & 15;
    const int col  = wave * 16 + (lane & 15);
    const int khi  = (lane >> 4) << 1;

    v8f acc = {};
#pragma unroll
    for (int k0 = 0; k0 < C; k0 += 4) {
        const int kk = k0 + khi;
        v2f a, w;
        a.x = sA[row * SMS + kk];
        a.y = sA[row * SMS + kk + 1];
        w.x = W1[kk * CH + col];
        w.y = W1[(kk + 1) * CH + col];
        acc = __builtin_amdgcn_wmma_f32_16x16x4_f32(false, a, false, w,
                                                    (short)0, acc, false, false);
    }
    const float scale = g[col] * __frsqrt_rn(v[col] + EPS);
    const float shift = bb[col] - m[col] * scale;
    const float bias  = b1[col];
    const int rbase   = node0 + ((lane >> 4) << 3);
#pragma unroll
    for (int j = 0; j < 8; ++j) {
        int r = rbase + j;
        float y = fmaf(acc[j] + bias, scale, shift);
        out[(size_t)r * CH + col] = fmaxf(y, 0.0f);
    }
}

// ---------------- head lin2 (64->1): one wave32 per node, shuffle reduce ----------------
__global__ __launch_bounds__(256) void head2_kernel(const float* __restrict__ t,
                                                    const float* __restrict__ W2,
                                                    const float* __restrict__ b2,
                                                    float* __restrict__ out, int n) {
    int gid  = blockIdx.x * blockDim.x + threadIdx.x;
    int node = gid >> 5;
    int lane = gid & 31;
    if (node >= n) return;
    float s = t[(size_t)node * CH + lane] * W2[lane]
            + t[(size_t)node * CH + lane + 32] * W2[lane + 32];
    for (int m = 16; m; m >>= 1) s += __shfl_xor(s, m, 32);
    if (lane == 0) out[node] = s + b2[0];
}

extern "C" void kernel_launch(void* const* d_in, const int* in_sizes, int n_in,
                              void* d_out, int out_size, void* d_ws, size_t ws_size,
                              hipStream_t stream) {
    const float*     x     = (const float*)d_in[0];
    const long long* ei    = (const long long*)d_in[1];
    const float*     Wl    = (const float*)d_in[2];
    const float*     Wr    = (const float*)d_in[3];
    const float*     bconv = (const float*)d_in[4];
    const float*     bn_g  = (const float*)d_in[5];
    const float*     bn_b  = (const float*)d_in[6];
    const float*     bn_m  = (const float*)d_in[7];
    const float*     bn_v  = (const float*)d_in[8];
    const float*     ln_g  = (const float*)d_in[9];
    const float*     ln_b  = (const float*)d_in[10];
    const float*     W1    = (const float*)d_in[11];
    const float*     b1    = (const float*)d_in[12];
    const float*     bo_g  = (const float*)d_in[13];
    const float*     bo_b  = (const float*)d_in[14];
    const float*     bo_m  = (const float*)d_in[15];
    const float*     bo_v  = (const float*)d_in[16];
    const float*     W2    = (const float*)d_in[17];
    const float*     b2    = (const float*)d_in[18];
    float*           out   = (float*)d_out;

    char* ws = (char*)d_ws;
    size_t off = 0;
    float* hA   = (float*)(ws + off); off += (size_t)N_NODES * C * sizeof(float);
    float* hB   = (float*)(ws + off); off += (size_t)N_NODES * C * sizeof(float);
    float* agg  = (float*)(ws + off); off += (size_t)N_NODES * C * sizeof(float);
    float* invc = (float*)(ws + off); off += (size_t)N_NODES * sizeof(float);
    float* t64  = (float*)(ws + off); off += (size_t)N_NODES * CH * sizeof(float);
    (void)off; (void)ws_size; (void)in_sizes; (void)n_in; (void)out_size;

    const long long* src = ei;             // edge_index[0]
    const long long* dst = ei + N_EDGES;   // edge_index[1]

    // in-degree -> inverse (dst fixed across layers: compute once)
    hipMemsetAsync(invc, 0, N_NODES * sizeof(float), stream);
    degree_kernel<<<(N_EDGES + 255) / 256, 256, 0, stream>>>(dst, invc, N_EDGES);
    inv_cnt_kernel<<<(N_NODES + 255) / 256, 256, 0, stream>>>(invc, N_NODES);

    const int gemm_blocks = (N_NODES + 16 * RT - 1) / (16 * RT);

    const float* cur = x;
    float*       nxt = hA;
    for (int i = 0; i < L; ++i) {
        hipMemsetAsync(agg, 0, (size_t)N_NODES * C * sizeof(float), stream);
        scatter_kernel<<<(int)(((size_t)N_EDGES * 32 + 255) / 256), 256, 0, stream>>>(
            cur, src, dst, agg, N_EDGES);
        sage_gemm_kernel<<<gemm_blocks, 256, 0, stream>>>(
            agg, invc, cur,
            Wl + (size_t)i * C * C, Wr + (size_t)i * C * C, bconv + i * C,
            bn_g + i * C, bn_b + i * C, bn_m + i * C, bn_v + i * C,
            (i == 0) ? x : nullptr, nxt);
        cur = nxt;
        nxt = (cur == hA) ? hB : hA;
    }

    // LayerNorm: cur -> nxt
    layernorm_kernel<<<(int)(((size_t)N_NODES * 32 + 255) / 256), 256, 0, stream>>>(
        cur, ln_g, ln_b, nxt, N_NODES);

    // MLP head
    head1_kernel<<<N_NODES / 16, 128, 0, stream>>>(nxt, W1, b1, bo_g, bo_b, bo_m, bo_v, t64);
    head2_kernel<<<(int)(((size_t)N_NODES * 32 + 255) / 256), 256, 0, stream>>>(
        t64, W2, b2, out, N_NODES);
}